// GATConvModule_74861279969842
// MI455X (gfx1250) — compile-verified
//
#include <hip/hip_runtime.h>
#include <math.h>

typedef float v2f __attribute__((ext_vector_type(2)));
typedef float v8f __attribute__((ext_vector_type(8)));

#define D 128
#define LDS_PITCH 132   // pad stride-128 rows to avoid 64-bank conflicts
#define NEG_SLOPE 0.2f

__device__ __forceinline__ float lrelu(float e) { return e >= 0.f ? e : NEG_SLOPE * e; }

// Monotone order-preserving float->uint key (works for negatives), so we can
// do segment-max with integer atomicMax.
__device__ __forceinline__ unsigned fkey(float f) {
  unsigned u = __float_as_uint(f);
  return (u & 0x80000000u) ? ~u : (u | 0x80000000u);
}
__device__ __forceinline__ float funkey(unsigned k) {
  unsigned u = (k & 0x80000000u) ? (k ^ 0x80000000u) : ~k;
  return __uint_as_float(u);
}

// ---------------------------------------------------------------------------
// K1: h = x @ W  via V_WMMA_F32_16X16X4_F32 (fp32 matrix core path).
// Block = 256 threads = 8 waves. Block b owns node rows [16b,16b+16);
// wave w owns output columns [16w,16w+16). A-tile staged in LDS (coalesced).
// ---------------------------------------------------------------------------
__global__ __launch_bounds__(256) void k_gemm(const float* __restrict__ x,
                                              const float* __restrict__ W,
                                              float* __restrict__ h, int N) {
  __shared__ float As[16 * LDS_PITCH];
  const int block_row = blockIdx.x * 16;
  const int tid = threadIdx.x;

  // Cooperative, coalesced load of the 16x128 A tile.
  for (int i = tid; i < 16 * D; i += 256) {
    int r = i >> 7;        // /128
    int c = i & (D - 1);
    int gr = block_row + r;
    As[r * LDS_PITCH + c] = (gr < N) ? x[(size_t)gr * D + c] : 0.0f;
  }
  __syncthreads();

  const int wave = tid >> 5;       // column tile 0..7
  const int lane = tid & 31;
  const int half = lane >> 4;      // K sub-pair select (0: K={k,k+1}, 1: K={k+2,k+3})
  const int l = lane & 15;
  const int n0 = wave * 16;

  v8f acc = {};
#pragma unroll
  for (int k = 0; k < D; k += 4) {
    v2f a, b;
    // A 16x4 fragment: lane l holds row M=l; VGPR0/1 = K pair selected by half.
    a.x = As[l * LDS_PITCH + k + 2 * half];
    a.y = As[l * LDS_PITCH + k + 1 + 2 * half];
    // B 4x16 fragment: lane l holds column N=n0+l; rows K striped like A.
    b.x = W[(size_t)(k + 2 * half) * D + n0 + l];
    b.y = W[(size_t)(k + 1 + 2 * half) * D + n0 + l];
    acc = __builtin_amdgcn_wmma_f32_16x16x4_f32(false, a, false, b,
                                                (short)0, acc, false, false);
  }

  // C/D layout: VGPR r -> (M=r, N=l) for lanes 0-15, (M=r+8, N=l) for lanes 16-31.
#pragma unroll
  for (int r = 0; r < 8; ++r) {
    int gr = block_row + r + 8 * half;
    if (gr < N) h[(size_t)gr * D + n0 + l] = acc[r];
  }
}

// ---------------------------------------------------------------------------
// K2: per-node attention logits a_src[n]=h[n]·att_src, a_dst[n]=h[n]·att_dst.
// One wave32 per node, 4 elements per lane, shuffle-tree reduce.
// ---------------------------------------------------------------------------
__global__ __launch_bounds__(256) void k_att(const float* __restrict__ h,
                                             const float* __restrict__ att_src,
                                             const float* __restrict__ att_dst,
                                             float* __restrict__ a_src,
                                             float* __restrict__ a_dst, int N) {
  int node = blockIdx.x * 8 + (threadIdx.x >> 5);
  int lane = threadIdx.x & 31;
  if (node >= N) return;
  const float* row = h + (size_t)node * D;
  float s = 0.f, d = 0.f;
#pragma unroll
  for (int i = 0; i < 4; ++i) {
    int c = lane + i * 32;
    float v = row[c];
    s += v * att_src[c];
    d += v * att_dst[c];
  }
#pragma unroll
  for (int off = 16; off > 0; off >>= 1) {
    s += __shfl_xor(s, off, 32);
    d += __shfl_xor(d, off, 32);
  }
  if (lane == 0) { a_src[node] = s; a_dst[node] = d; }
}

// K3: seed segment-max with the self-loop logit (encoded as uint key).
__global__ void k_init_max(const float* __restrict__ a_src,
                           const float* __restrict__ a_dst,
                           unsigned* __restrict__ m_enc, int N) {
  int n = blockIdx.x * blockDim.x + threadIdx.x;
  if (n < N) m_enc[n] = fkey(lrelu(a_src[n] + a_dst[n]));
}

// K4: per-edge atomicMax into destination's running max.
__global__ void k_edge_max(const int* __restrict__ ei,
                           const float* __restrict__ a_src,
                           const float* __restrict__ a_dst,
                           unsigned* __restrict__ m_enc, int E) {
  int e = blockIdx.x * blockDim.x + threadIdx.x;
  if (e >= E) return;
  int s = ei[e], d = ei[E + e];
  atomicMax(&m_enc[d], fkey(lrelu(a_src[s] + a_dst[d])));
}

// K5: decode max, seed denominator with self-loop exp term.
__global__ void k_den_init(const float* __restrict__ a_src,
                           const float* __restrict__ a_dst,
                           const unsigned* __restrict__ m_enc,
                           float* __restrict__ m, float* __restrict__ denom,
                           float* __restrict__ ex_self, int N) {
  int n = blockIdx.x * blockDim.x + threadIdx.x;
  if (n >= N) return;
  float mm = funkey(m_enc[n]);
  m[n] = mm;
  float ex = __expf(lrelu(a_src[n] + a_dst[n]) - mm);
  ex_self[n] = ex;
  denom[n] = ex;
}

// K6: per-edge exp term; cache it and accumulate denominator.
__global__ void k_edge_den(const int* __restrict__ ei,
                           const float* __restrict__ a_src,
                           const float* __restrict__ a_dst,
                           const float* __restrict__ m,
                           float* __restrict__ denom,
                           float* __restrict__ ex_arr, int E) {
  int e = blockIdx.x * blockDim.x + threadIdx.x;
  if (e >= E) return;
  int s = ei[e], d = ei[E + e];
  float ex = __expf(lrelu(a_src[s] + a_dst[d]) - m[d]);
  ex_arr[e] = ex;
  atomicAdd(&denom[d], ex);
}

// K7: out[n,:] = bias + alpha_self * h[n,:]  (fully overwrites poisoned d_out).
__global__ void k_self_agg(const float* __restrict__ h,
                           const float* __restrict__ ex_self,
                           const float* __restrict__ denom,
                           const float* __restrict__ bias,
                           float* __restrict__ out, int N) {
  int i = blockIdx.x * blockDim.x + threadIdx.x;
  if (i >= N * D) return;
  int n = i >> 7;
  int c = i & (D - 1);
  out[i] = bias[c] + h[i] * (ex_self[n] / denom[n]);
}

// K8: per-edge scatter: out[dst,:] += alpha_e * h[src,:].
// One wave32 per edge; float4 gather of h[src] (512B/wave, coalesced),
// 4 fp32 atomics per lane into L2-resident out.
__global__ __launch_bounds__(256) void k_edge_agg(const int* __restrict__ ei,
                                                  const float* __restrict__ h,
                                                  const float* __restrict__ ex_arr,
                                                  const float* __restrict__ denom,
                                                  float* __restrict__ out, int E) {
  int e = blockIdx.x * 8 + (threadIdx.x >> 5);
  int lane = threadIdx.x & 31;
  if (e >= E) return;
  int s = ei[e], d = ei[E + e];
  float alpha = ex_arr[e] / denom[d];
  const float4 v = ((const float4*)(h + (size_t)s * D))[lane];
  float* od = out + (size_t)d * D + lane * 4;
  atomicAdd(od + 0, v.x * alpha);
  atomicAdd(od + 1, v.y * alpha);
  atomicAdd(od + 2, v.z * alpha);
  atomicAdd(od + 3, v.w * alpha);
}

extern "C" void kernel_launch(void* const* d_in, const int* in_sizes, int n_in,
                              void* d_out, int out_size, void* d_ws, size_t ws_size,
                              hipStream_t stream) {
  const float* x       = (const float*)d_in[0];
  const int*   ei      = (const int*)d_in[1];
  const float* W       = (const float*)d_in[2];
  const float* att_src = (const float*)d_in[3];
  const float* att_dst = (const float*)d_in[4];
  const float* bias    = (const float*)d_in[5];
  float* out = (float*)d_out;

  const int N = in_sizes[0] / D;
  const int E = in_sizes[1] / 2;

  // Workspace layout (all 4-byte elements).
  float*    h       = (float*)d_ws;
  float*    a_src   = h + (size_t)N * D;
  float*    a_dst   = a_src + N;
  unsigned* m_enc   = (unsigned*)(a_dst + N);
  float*    m       = (float*)(m_enc + N);
  float*    denom   = m + N;
  float*    ex_self = denom + N;
  float*    ex_arr  = ex_self + N;

  const int row_blocks  = (N + 15) / 16;
  const int node_waves  = (N + 7) / 8;
  const int node_thr    = (N + 255) / 256;
  const int edge_thr    = (E + 255) / 256;
  const int elem_thr    = (N * D + 255) / 256;
  const int edge_waves  = (E + 7) / 8;

  k_gemm<<<row_blocks, 256, 0, stream>>>(x, W, h, N);
  k_att<<<node_waves, 256, 0, stream>>>(h, att_src, att_dst, a_src, a_dst, N);
  k_init_max<<<node_thr, 256, 0, stream>>>(a_src, a_dst, m_enc, N);
  k_edge_max<<<edge_thr, 256, 0, stream>>>(ei, a_src, a_dst, m_enc, E);
  k_den_init<<<node_thr, 256, 0, stream>>>(a_src, a_dst, m_enc, m, denom, ex_self, N);
  k_edge_den<<<edge_thr, 256, 0, stream>>>(ei, a_src, a_dst, m, denom, ex_arr, E);
  k_self_agg<<<elem_thr, 256, 0, stream>>>(h, ex_self, denom, bias, out, N);
  k_edge_agg<<<edge_waves, 256, 0, stream>>>(ei, h, ex_arr, denom, out, E);
}